// ComplexCNN_90855738179624
// MI455X (gfx1250) — compile-verified
//
#include <hip/hip_runtime.h>
#include <hip/hip_bf16.h>
#include <math.h>

// ---------------------------------------------------------------------------
// CDNA5 (gfx1250) complex CNN forward pass.
// Heavy ops (convs as implicit GEMM, FC layers) run on V_WMMA_F32_16X16X4_F32.
// One wave32 owns a 32x16 output tile: two 16x16 WMMA tiles sharing the same
// B fragment (8 WMMA accumulations per K-step of 4).  Conv patch offsets are
// precomputed into an LDS table to keep div/mod out of the inner loop.
// ---------------------------------------------------------------------------

typedef __attribute__((ext_vector_type(2))) float v2f;
typedef __attribute__((ext_vector_type(8))) float v8f;

__device__ __forceinline__ v8f wmma4(v2f a, v2f b, v8f c) {
    // 8 args: (neg_a, A, neg_b, B, c_mod, C, reuse_a, reuse_b)
    return __builtin_amdgcn_wmma_f32_16x16x4_f32(false, a, false, b, (short)0, c,
                                                 false, false);
}

// ---------------------------------------------------------------------------
// Complex conv as implicit GEMM.
//   A (2 x 16x4) = weights, rows = output channels (two 16-row tiles)
//   B (4x16)     = input patches, cols = output pixels (n, oh, ow)
// grid.x = Npix/16, grid.y = Cout/32, block = 32 (one wave).
// ---------------------------------------------------------------------------
__global__ void cconv_wmma_kernel(
    const float* __restrict__ xr, const float* __restrict__ xi,
    const float* __restrict__ wr, const float* __restrict__ wi,
    const float* __restrict__ br, const float* __restrict__ bi,
    float* __restrict__ yr, float* __restrict__ yi,
    int Cin, int Hi, int Wi, int Cout, int Ho, int Wo)
{
    const int lane = threadIdx.x;      // 0..31
    const int half = lane >> 4;        // 0 or 1
    const int l15  = lane & 15;
    const int K    = Cin * 9;
    const int HoWo = Ho * Wo;
    const int koff = half * 2;

    // Precompute im2col offsets into LDS (once per block): offs[k] for k<K.
    __shared__ int offs[576];          // max K = 64*9
    for (int k = lane; k < K; k += 32) {
        int ci = k / 9; int r9 = k - ci * 9;
        int r = r9 / 3; int s = r9 - r * 3;
        offs[k] = (ci * Hi + r) * Wi + s;
    }
    __syncthreads();

    // B-matrix column handled by this lane (output pixel)
    const int col   = blockIdx.x * 16 + l15;
    const int n_img = col / HoWo;
    const int rem   = col - n_img * HoWo;
    const int oh    = rem / Wo;
    const int ow    = rem - oh * Wo;
    const int in_base = (n_img * Cin) * Hi * Wi + oh * Wi + ow;

    // A-matrix rows handled by this lane (two output-channel tiles)
    const int m0 = blockIdx.y * 32 + l15;
    const int m1 = m0 + 16;
    const float* wr0 = wr + (size_t)m0 * K;
    const float* wi0 = wi + (size_t)m0 * K;
    const float* wr1 = wr + (size_t)m1 * K;
    const float* wi1 = wi + (size_t)m1 * K;

    v8f accr0 = {}, acci0 = {}, accr1 = {}, acci1 = {};

    for (int k0 = 0; k0 < K; k0 += 4) {
        const int ka = k0 + koff;
        const bool g0 = (ka     < K);
        const bool g1 = (ka + 1 < K);

        // A fragments (weights), zero-padded past K (conv1 has K=9)
        v2f ar0, ai0, ar1, ai1;
        ar0.x = g0 ? wr0[ka] : 0.0f;  ar0.y = g1 ? wr0[ka + 1] : 0.0f;
        ai0.x = g0 ? wi0[ka] : 0.0f;  ai0.y = g1 ? wi0[ka + 1] : 0.0f;
        ar1.x = g0 ? wr1[ka] : 0.0f;  ar1.y = g1 ? wr1[ka + 1] : 0.0f;
        ai1.x = g0 ? wi1[ka] : 0.0f;  ai1.y = g1 ? wi1[ka + 1] : 0.0f;

        // B fragments (patches): LDS offset table + one add per element
        float pr0 = 0.0f, pr1 = 0.0f, pi0 = 0.0f, pi1 = 0.0f;
        if (g0) { int a = in_base + offs[ka];     pr0 = xr[a]; pi0 = xi[a]; }
        if (g1) { int a = in_base + offs[ka + 1]; pr1 = xr[a]; pi1 = xi[a]; }

        v2f b_r  = { pr0,  pr1 };
        v2f b_i  = { pi0,  pi1 };
        v2f b_ni = { -pi0, -pi1 };

        // yr += Wr*Xr - Wi*Xi ; yi += Wi*Xr + Wr*Xi   (both M-tiles share B)
        accr0 = wmma4(ar0, b_r,  accr0);
        accr0 = wmma4(ai0, b_ni, accr0);
        acci0 = wmma4(ai0, b_r,  acci0);
        acci0 = wmma4(ar0, b_i,  acci0);
        accr1 = wmma4(ar1, b_r,  accr1);
        accr1 = wmma4(ai1, b_ni, accr1);
        acci1 = wmma4(ai1, b_r,  acci1);
        acci1 = wmma4(ar1, b_i,  acci1);
    }

    // C/D layout: VGPR v <-> row (v + 8*half), col = lane&15
#pragma unroll
    for (int v = 0; v < 8; ++v) {
        int m = blockIdx.y * 32 + v + half * 8;
        size_t o = ((size_t)(n_img * Cout + m) * Ho + oh) * Wo + ow;
        yr[o] = accr0[v] + br[m];
        yi[o] = acci0[v] + bi[m];
        int m2 = m + 16;
        size_t o2 = ((size_t)(n_img * Cout + m2) * Ho + oh) * Wo + ow;
        yr[o2] = accr1[v] + br[m2];
        yi[o2] = acci1[v] + bi[m2];
    }
}

// ---------------------------------------------------------------------------
// Complex FC:  Y[m][n] = sum_k X[m][k] * W[n][k]  (complex), bias fused.
// M is fixed at 32 (two 16-row tiles per wave).  grid.x = ceil(N/16).
// ---------------------------------------------------------------------------
__global__ void cgemm_fc_kernel(
    const float* __restrict__ xr, const float* __restrict__ xi,
    const float* __restrict__ wr, const float* __restrict__ wi,
    const float* __restrict__ br, const float* __restrict__ bi,
    float* __restrict__ yr, float* __restrict__ yi,
    int N, int K)
{
    const int lane = threadIdx.x;
    const int half = lane >> 4;
    const int l15  = lane & 15;
    const int koff = half * 2;

    const int ncol = blockIdx.x * 16 + l15;
    const bool nok = (ncol < N);

    const float* xr0 = xr + (size_t)l15 * K;
    const float* xi0 = xi + (size_t)l15 * K;
    const float* xr1 = xr + (size_t)(l15 + 16) * K;
    const float* xi1 = xi + (size_t)(l15 + 16) * K;
    const float* wr_row = wr + (size_t)(nok ? ncol : 0) * K;
    const float* wi_row = wi + (size_t)(nok ? ncol : 0) * K;

    v8f accr0 = {}, acci0 = {}, accr1 = {}, acci1 = {};

    for (int k0 = 0; k0 < K; k0 += 4) {  // K is a multiple of 4 for all FCs
        const int ka = k0 + koff;
        v2f ar0 = { xr0[ka], xr0[ka + 1] };
        v2f ai0 = { xi0[ka], xi0[ka + 1] };
        v2f ar1 = { xr1[ka], xr1[ka + 1] };
        v2f ai1 = { xi1[ka], xi1[ka + 1] };

        float b0r = nok ? wr_row[ka]     : 0.0f;
        float b1r = nok ? wr_row[ka + 1] : 0.0f;
        float b0i = nok ? wi_row[ka]     : 0.0f;
        float b1i = nok ? wi_row[ka + 1] : 0.0f;
        v2f b_r  = { b0r,  b1r };
        v2f b_i  = { b0i,  b1i };
        v2f b_ni = { -b0i, -b1i };

        accr0 = wmma4(ar0, b_r,  accr0);
        accr0 = wmma4(ai0, b_ni, accr0);
        acci0 = wmma4(ai0, b_r,  acci0);
        acci0 = wmma4(ar0, b_i,  acci0);
        accr1 = wmma4(ar1, b_r,  accr1);
        accr1 = wmma4(ai1, b_ni, accr1);
        acci1 = wmma4(ai1, b_r,  acci1);
        acci1 = wmma4(ar1, b_i,  acci1);
    }

#pragma unroll
    for (int v = 0; v < 8; ++v) {
        if (nok) {
            int m  = v + half * 8;
            int m2 = m + 16;
            yr[(size_t)m  * N + ncol] = accr0[v] + br[ncol];
            yi[(size_t)m  * N + ncol] = acci0[v] + bi[ncol];
            yr[(size_t)m2 * N + ncol] = accr1[v] + br[ncol];
            yi[(size_t)m2 * N + ncol] = acci1[v] + bi[ncol];
        }
    }
}

// ---------------------------------------------------------------------------
// Complex batch-norm: per-channel 5 moments -> whitening coeffs -> apply+ReLU
// ---------------------------------------------------------------------------
__global__ void bn_zero_kernel(float* s, int n)
{
    int i = blockIdx.x * blockDim.x + threadIdx.x;
    if (i < n) s[i] = 0.0f;
}

__global__ void bn_reduce_kernel(const float* __restrict__ xr,
                                 const float* __restrict__ xi,
                                 float* __restrict__ sums,
                                 int C, int HW, int Nb)
{
    const int c      = blockIdx.x;
    const int split  = blockIdx.y;
    const int nsplit = gridDim.y;
    const int tid    = threadIdx.x;

    float sr = 0.f, si = 0.f, srr = 0.f, sii = 0.f, sri = 0.f;
    for (int n = 0; n < Nb; ++n) {
        const float* xrb = xr + ((size_t)n * C + c) * HW;
        const float* xib = xi + ((size_t)n * C + c) * HW;
        for (int j = split * blockDim.x + tid; j < HW;
             j += nsplit * blockDim.x) {
            float r = xrb[j], im = xib[j];
            sr += r; si += im; srr += r * r; sii += im * im; sri += r * im;
        }
    }

    __shared__ float sm[5][256];
    sm[0][tid] = sr; sm[1][tid] = si; sm[2][tid] = srr;
    sm[3][tid] = sii; sm[4][tid] = sri;
    __syncthreads();
    for (int s = 128; s > 0; s >>= 1) {
        if (tid < s) {
#pragma unroll
            for (int q = 0; q < 5; ++q) sm[q][tid] += sm[q][tid + s];
        }
        __syncthreads();
    }
    if (tid == 0) {
#pragma unroll
        for (int q = 0; q < 5; ++q) atomicAdd(&sums[c * 5 + q], sm[q][0]);
    }
}

__global__ void bn_coef_kernel(const float* __restrict__ sums,
                               const float* __restrict__ bnw,
                               const float* __restrict__ bnb,
                               float* __restrict__ coef, int C, float inv_cnt)
{
    int c = blockIdx.x * blockDim.x + threadIdx.x;
    if (c >= C) return;
    float mr  = sums[c * 5 + 0] * inv_cnt;
    float mi  = sums[c * 5 + 1] * inv_cnt;
    float Vrr = sums[c * 5 + 2] * inv_cnt - mr * mr + 1e-5f;
    float Vii = sums[c * 5 + 3] * inv_cnt - mi * mi + 1e-5f;
    float Vri = sums[c * 5 + 4] * inv_cnt - mr * mi;
    float s      = sqrtf(Vrr * Vii - Vri * Vri);
    float t      = sqrtf(Vrr + Vii + 2.0f * s);
    float inv_st = 1.0f / (s * t);
    float Rrr = (Vii + s) * inv_st;
    float Rii = (Vrr + s) * inv_st;
    float Rri = -Vri * inv_st;
    float Wrr = bnw[c * 3 + 0], Wii = bnw[c * 3 + 1], Wri = bnw[c * 3 + 2];
    coef[c * 8 + 0] = mr;
    coef[c * 8 + 1] = mi;
    coef[c * 8 + 2] = Wrr * Rrr + Wri * Rri;   // a_rr
    coef[c * 8 + 3] = Wrr * Rri + Wri * Rii;   // a_ri
    coef[c * 8 + 4] = Wri * Rrr + Wii * Rri;   // a_ir
    coef[c * 8 + 5] = Wri * Rri + Wii * Rii;   // a_ii
    coef[c * 8 + 6] = bnb[c * 2 + 0];
    coef[c * 8 + 7] = bnb[c * 2 + 1];
}

__global__ void bn_apply_relu_kernel(float* __restrict__ xr, float* __restrict__ xi,
                                     const float* __restrict__ coef,
                                     int C, int HW, long total)
{
    long i = (long)blockIdx.x * blockDim.x + threadIdx.x;
    if (i >= total) return;
    int c = (int)((i / HW) % C);
    const float* cf = coef + c * 8;
    float cr = xr[i] - cf[0];
    float ci = xi[i] - cf[1];
    float orv = cf[2] * cr + cf[3] * ci + cf[6];
    float oiv = cf[4] * cr + cf[5] * ci + cf[7];
    xr[i] = orv > 0.0f ? orv : 0.0f;
    xi[i] = oiv > 0.0f ? oiv : 0.0f;
}

// ---------------------------------------------------------------------------
// Magnitude-argmax 2x2 pool (first max wins, as jnp.argmax)
// ---------------------------------------------------------------------------
__global__ void cpool_kernel(const float* __restrict__ xr, const float* __restrict__ xi,
                             float* __restrict__ yr, float* __restrict__ yi,
                             int C, int H, int W, int H2, int W2, long total)
{
    long i = (long)blockIdx.x * blockDim.x + threadIdx.x;
    if (i >= total) return;
    int w2 = (int)(i % W2); long t = i / W2;
    int h2 = (int)(t % H2); t /= H2;
    int c  = (int)(t % C);
    int n  = (int)(t / C);
    size_t base = ((size_t)(n * C + c) * H + h2 * 2) * W + w2 * 2;

    float best = -1.0f, br_ = 0.0f, bi_ = 0.0f;
#pragma unroll
    for (int dy = 0; dy < 2; ++dy) {
#pragma unroll
        for (int dx = 0; dx < 2; ++dx) {
            size_t a = base + (size_t)dy * W + dx;
            float r = xr[a], im = xi[a];
            float m = r * r + im * im;
            if (m > best) { best = m; br_ = r; bi_ = im; }
        }
    }
    yr[i] = br_;
    yi[i] = bi_;
}

// ---------------------------------------------------------------------------
// log_softmax(hr^2 + hi^2) per row
// ---------------------------------------------------------------------------
__global__ void logsoftmax_kernel(const float* __restrict__ hr,
                                  const float* __restrict__ hi,
                                  float* __restrict__ out, int B, int NC)
{
    int b = blockIdx.x * blockDim.x + threadIdx.x;
    if (b >= B) return;
    float p[16];
    float mx = -3.4e38f;
    for (int j = 0; j < NC; ++j) {
        float r = hr[b * NC + j], im = hi[b * NC + j];
        p[j] = r * r + im * im;
        mx = p[j] > mx ? p[j] : mx;
    }
    float s = 0.0f;
    for (int j = 0; j < NC; ++j) s += expf(p[j] - mx);
    float l = logf(s);
    for (int j = 0; j < NC; ++j) out[b * NC + j] = p[j] - mx - l;
}

// ---------------------------------------------------------------------------
// Host-side orchestration
// ---------------------------------------------------------------------------
static void run_bn(float* xr, float* xi, const float* bnw, const float* bnb,
                   float* sums, float* coef, int C, int HW, int Nb,
                   hipStream_t stream)
{
    long count = (long)Nb * HW;
    bn_zero_kernel<<<dim3((5 * C + 255) / 256), dim3(256), 0, stream>>>(sums, 5 * C);
    long ns = count / (256 * 16);
    if (ns < 1) ns = 1;
    if (ns > 64) ns = 64;
    bn_reduce_kernel<<<dim3(C, (unsigned)ns), dim3(256), 0, stream>>>(xr, xi, sums, C, HW, Nb);
    bn_coef_kernel<<<dim3((C + 255) / 256), dim3(256), 0, stream>>>(
        sums, bnw, bnb, coef, C, 1.0f / (float)count);
    long total = (long)C * count;
    bn_apply_relu_kernel<<<dim3((unsigned)((total + 255) / 256)), dim3(256), 0, stream>>>(
        xr, xi, coef, C, HW, total);
}

extern "C" void kernel_launch(void* const* d_in, const int* in_sizes, int n_in,
                              void* d_out, int out_size, void* d_ws, size_t ws_size,
                              hipStream_t stream)
{
    const float* x_r  = (const float*)d_in[0];
    const float* x_i  = (const float*)d_in[1];
    const float* c1wr = (const float*)d_in[2];
    const float* c1wi = (const float*)d_in[3];
    const float* c1br = (const float*)d_in[4];
    const float* c1bi = (const float*)d_in[5];
    const float* c2wr = (const float*)d_in[6];
    const float* c2wi = (const float*)d_in[7];
    const float* c2br = (const float*)d_in[8];
    const float* c2bi = (const float*)d_in[9];
    const float* c3wr = (const float*)d_in[10];
    const float* c3wi = (const float*)d_in[11];
    const float* c3br = (const float*)d_in[12];
    const float* c3bi = (const float*)d_in[13];
    const float* bn1w = (const float*)d_in[14];
    const float* bn1b = (const float*)d_in[15];
    const float* bn2w = (const float*)d_in[16];
    const float* bn2b = (const float*)d_in[17];
    const float* bn3w = (const float*)d_in[18];
    const float* bn3b = (const float*)d_in[19];
    const float* bn4w = (const float*)d_in[20];
    const float* bn4b = (const float*)d_in[21];
    const float* bn5w = (const float*)d_in[22];
    const float* bn5b = (const float*)d_in[23];
    const float* f1wr = (const float*)d_in[24];
    const float* f1wi = (const float*)d_in[25];
    const float* f1br = (const float*)d_in[26];
    const float* f1bi = (const float*)d_in[27];
    const float* f2wr = (const float*)d_in[28];
    const float* f2wi = (const float*)d_in[29];
    const float* f2br = (const float*)d_in[30];
    const float* f2bi = (const float*)d_in[31];
    const float* cwr  = (const float*)d_in[32];
    const float* cwi  = (const float*)d_in[33];
    const float* cbr  = (const float*)d_in[34];
    const float* cbi  = (const float*)d_in[35];

    float* ws = (float*)d_ws;
    // workspace layout (floats)
    float* c_r  = ws;                      // conv/FC buffer real  (16,777,216)
    float* c_i  = ws + 16777216;           // conv/FC buffer imag
    float* p_r  = ws + 33554432;           // pooled buffer real   ( 4,194,304)
    float* p_i  = ws + 37748736;           // pooled buffer imag
    float* sums = ws + 41943040;           // 5*512
    float* coef = sums + 5 * 512;          // 8*512

    const dim3 b32(32), b256(256);
    const int B = 32;

    // ---- stage 1: conv1 (1->32, 130->128), bn1, relu, pool -> 64x64
    cconv_wmma_kernel<<<dim3(524288 / 16, 32 / 32), b32, 0, stream>>>(
        x_r, x_i, c1wr, c1wi, c1br, c1bi, c_r, c_i, 1, 130, 130, 32, 128, 128);
    run_bn(c_r, c_i, bn1w, bn1b, sums, coef, 32, 128 * 128, B, stream);
    {
        long total = (long)B * 32 * 64 * 64;
        cpool_kernel<<<dim3((unsigned)((total + 255) / 256)), b256, 0, stream>>>(
            c_r, c_i, p_r, p_i, 32, 128, 128, 64, 64, total);
    }

    // ---- stage 2: conv2 (32->64, 64->62), bn2, relu, pool -> 31x31
    cconv_wmma_kernel<<<dim3(123008 / 16, 64 / 32), b32, 0, stream>>>(
        p_r, p_i, c2wr, c2wi, c2br, c2bi, c_r, c_i, 32, 64, 64, 64, 62, 62);
    run_bn(c_r, c_i, bn2w, bn2b, sums, coef, 64, 62 * 62, B, stream);
    {
        long total = (long)B * 64 * 31 * 31;
        cpool_kernel<<<dim3((unsigned)((total + 255) / 256)), b256, 0, stream>>>(
            c_r, c_i, p_r, p_i, 64, 62, 62, 31, 31, total);
    }

    // ---- stage 3: conv3 (64->128, 31->29), bn3, relu, pool -> 14x14
    cconv_wmma_kernel<<<dim3(26912 / 16, 128 / 32), b32, 0, stream>>>(
        p_r, p_i, c3wr, c3wi, c3br, c3bi, c_r, c_i, 64, 31, 31, 128, 29, 29);
    run_bn(c_r, c_i, bn3w, bn3b, sums, coef, 128, 29 * 29, B, stream);
    {
        long total = (long)B * 128 * 14 * 14;   // p_r/p_i = flattened (32, 25088)
        cpool_kernel<<<dim3((unsigned)((total + 255) / 256)), b256, 0, stream>>>(
            c_r, c_i, p_r, p_i, 128, 29, 29, 14, 14, total);
    }

    // ---- FC1: (32 x 25088) -> (32 x 512), bn4, relu
    cgemm_fc_kernel<<<dim3(512 / 16, 1), b32, 0, stream>>>(
        p_r, p_i, f1wr, f1wi, f1br, f1bi, c_r, c_i, 512, 25088);
    run_bn(c_r, c_i, bn4w, bn4b, sums, coef, 512, 1, B, stream);

    // ---- FC2: (32 x 512) -> (32 x 512), bn5, relu
    cgemm_fc_kernel<<<dim3(512 / 16, 1), b32, 0, stream>>>(
        c_r, c_i, f2wr, f2wi, f2br, f2bi, p_r, p_i, 512, 512);
    run_bn(p_r, p_i, bn5w, bn5b, sums, coef, 512, 1, B, stream);

    // ---- classifier: (32 x 512) -> (32 x 10)
    cgemm_fc_kernel<<<dim3(1, 1), b32, 0, stream>>>(
        p_r, p_i, cwr, cwi, cbr, cbi, c_r, c_i, 10, 512);

    // ---- log_softmax(|h|^2)
    logsoftmax_kernel<<<dim3(1), dim3(32), 0, stream>>>(
        c_r, c_i, (float*)d_out, B, 10);
}